// InputEmbedderMultimer_23124103921939
// MI455X (gfx1250) — compile-verified
//
#include <hip/hip_runtime.h>
#include <stdint.h>

// ---------------------------------------------------------------------------
// Sizes (fixed by the reference)
// ---------------------------------------------------------------------------
#define NRES 768
#define SROW 512
#define TFD  21
#define MSAD 49
#define CZ   128
#define CM   256
#define NO_BINS 73          // 66 + 1 + 6
#define MAXR 32
#define MAXC 2

typedef __attribute__((ext_vector_type(2))) float v2f;
typedef __attribute__((ext_vector_type(4))) float f32x4;
typedef __attribute__((ext_vector_type(8))) float v8f;

__device__ __forceinline__ f32x4 ld4(const float* p) { return *(const f32x4*)p; }
__device__ __forceinline__ void st4nt(float* p, f32x4 v) {
  __builtin_nontemporal_store(v, (f32x4*)p);
}

// CDNA5 async copy: global -> LDS, tracked by ASYNCcnt (GV mode, saddr=off).
// LDS byte address = low 32 bits of the generic pointer to a __shared__ object.
__device__ __forceinline__ void async_ld_b32(uint32_t lds_addr, const float* g) {
  asm volatile("global_load_async_to_lds_b32 %0, %1, off"
               :: "v"(lds_addr), "v"(g) : "memory");
}
__device__ __forceinline__ void wait_async0() {
  asm volatile("s_wait_asynccnt 0" ::: "memory");
}

// ---------------------------------------------------------------------------
// Kernel 0: tiny projections  tf_emb_i / tf_emb_j / tf_m  (K = 21 dots)
//   ws layout (floats): tfi[768*128] @0, tfj[768*128] @98304, tfm[768*256] @196608
// ---------------------------------------------------------------------------
__global__ __launch_bounds__(256) void tf_proj_kernel(
    const float* __restrict__ tf,
    const float* __restrict__ wzi, const float* __restrict__ bzi,
    const float* __restrict__ wzj, const float* __restrict__ bzj,
    const float* __restrict__ wm,  const float* __restrict__ bm,
    float* __restrict__ ws)
{
  int t = blockIdx.x * 256 + threadIdx.x;      // 768*512 threads total
  int n = t >> 9;
  int c = t & 511;
  const float* trow = tf + n * TFD;
  if (c < CZ) {
    float a = bzi[c];
    #pragma unroll
    for (int k = 0; k < TFD; ++k) a = fmaf(trow[k], wzi[k * CZ + c], a);
    ws[n * CZ + c] = a;
  } else if (c < 2 * CZ) {
    int cc = c - CZ;
    float a = bzj[cc];
    #pragma unroll
    for (int k = 0; k < TFD; ++k) a = fmaf(trow[k], wzj[k * CZ + cc], a);
    ws[NRES * CZ + n * CZ + cc] = a;
  } else {
    int cc = c - 2 * CZ;
    float a = bm[cc];
    #pragma unroll
    for (int k = 0; k < TFD; ++k) a = fmaf(trow[k], wm[k * CM + cc], a);
    ws[2 * NRES * CZ + n * CM + cc] = a;
  }
}

// ---------------------------------------------------------------------------
// Kernel 1: msa_emb = msa_feat[S*N,49] @ w_msa[49,256] + b_msa + tf_m[n]
//   fp32 WMMA 16x16x4, K padded 49->52 (13 steps, 26 K-pairs).
//   Block = 256 threads (8 waves), 8 row-tiles of 16 (128 rows) per block.
//   Wave w owns output col tiles {32w, 32w+16}.
//   B in LDS as K-paired float2 (one ds_load_b64 per operand fetch).
//   A double-buffered in LDS, filled by global_load_async_to_lds_b32.
// ---------------------------------------------------------------------------
#define MT    8           // 16-row tiles per block
#define BK    52          // padded K (A row stride, floats)
#define NKP   26          // K pairs
#define B2STR 272         // Bs row stride in float2: 272*8B -> half-wave bank
                          // shift = 32 banks; lanes 0-15 use banks 0-31,
                          // lanes 16-31 use banks 32-63 (conflict-free)

__global__ __launch_bounds__(256) void msa_wmma_kernel(
    const float* __restrict__ msa_feat,
    const float* __restrict__ w_msa, const float* __restrict__ b_msa,
    const float* __restrict__ tfm,   float* __restrict__ out)
{
  __shared__ v2f   Bs[NKP * B2STR];        // 56,576 B
  __shared__ float As[2][16 * BK];         //  6,656 B (double-buffered A)

  const int tid  = threadIdx.x;
  const int wave = tid >> 5;
  const int lane = tid & 31;
  const int laneN  = lane & 15;
  const int laneHi = lane >> 4;            // 0 or 1

  // ---- stage B panel as K-pairs: Bs[kp][n] = {W[2kp][n], W[2kp+1][n]} ----
  for (int idx = tid; idx < NKP * CM; idx += 256) {
    int kp = idx >> 8;           // /256
    int n  = idx & 255;
    int k0 = kp * 2;
    v2f v;
    v.x = (k0     < MSAD) ? w_msa[k0 * CM + n]       : 0.0f;
    v.y = (k0 + 1 < MSAD) ? w_msa[(k0 + 1) * CM + n] : 0.0f;
    Bs[kp * B2STR + n] = v;
  }
  // ---- zero A pad columns (k=49..51) once; async copies never touch them ----
  if (tid < 2 * 16 * 3) {
    int b = tid / 48, r = tid % 48;
    int m = r / 3, k = MSAD + (r % 3);
    As[b][m * BK + k] = 0.0f;
  }

  const long blockRow = (long)blockIdx.x * (MT * 16);
  const int  nBase    = (int)(blockIdx.x % (NRES / (MT * 16))) * (MT * 16);

  // ---- prefetch A tile 0 (async, per-lane b32 copies) ----
  for (int idx = tid; idx < 16 * MSAD; idx += 256) {
    int m = idx / MSAD, k = idx - m * MSAD;
    async_ld_b32((uint32_t)(uintptr_t)&As[0][m * BK + k],
                 msa_feat + (blockRow + m) * MSAD + k);
  }
  wait_async0();
  __syncthreads();

  const int col0 = wave * 32 + laneN;
  const int col1 = col0 + 16;
  const float bias0 = b_msa[col0];
  const float bias1 = b_msa[col1];

  for (int t = 0; t < MT; ++t) {
    const float* A = As[t & 1];

    // ---- async-prefetch next A tile into the other buffer ----
    if (t + 1 < MT) {
      float* An = As[(t + 1) & 1];
      long rb = blockRow + (long)(t + 1) * 16;
      for (int idx = tid; idx < 16 * MSAD; idx += 256) {
        int m = idx / MSAD, k = idx - m * MSAD;
        async_ld_b32((uint32_t)(uintptr_t)&An[m * BK + k],
                     msa_feat + (rb + m) * MSAD + k);
      }
    }

    // ---- 13 fp32 WMMA steps over K ----
    v8f acc0 = {};
    v8f acc1 = {};
    #pragma unroll
    for (int kk = 0; kk < 13; ++kk) {
      const int k0 = kk * 4 + 2 * laneHi;     // A: lanes 0-15 K{0,1}, 16-31 K{2,3}
      v2f a;
      a.x = A[laneN * BK + k0];
      a.y = A[laneN * BK + k0 + 1];
      const int kp = kk * 2 + laneHi;         // B pair row (one ds_load_b64 each)
      v2f b0 = Bs[kp * B2STR + col0];
      v2f b1 = Bs[kp * B2STR + col1];
      acc0 = __builtin_amdgcn_wmma_f32_16x16x4_f32(
                false, a, false, b0, (short)0, acc0, false, false);
      acc1 = __builtin_amdgcn_wmma_f32_16x16x4_f32(
                false, a, false, b1, (short)0, acc1, false, false);
    }

    // ---- epilogue: + bias + tf_m[row % N], nontemporal store ----
    const int rowT = t * 16;
    #pragma unroll
    for (int v = 0; v < 8; ++v) {
      const int  rl = v + 8 * laneHi;         // C/D: lanes 0-15 M=v, 16-31 M=v+8
      const long g  = blockRow + rowT + rl;
      const int  n  = nBase + rowT + rl;      // == g % NRES (768 = 6*128)
      const float t0 = tfm[n * CM + col0];
      const float t1 = tfm[n * CM + col1];
      __builtin_nontemporal_store(acc0[v] + bias0 + t0, out + g * CM + col0);
      __builtin_nontemporal_store(acc1[v] + bias1 + t1, out + g * CM + col1);
    }

    wait_async0();        // next tile's async copies complete
    __syncthreads();      // visible to all waves; buffer swap safe
  }
}

// ---------------------------------------------------------------------------
// Kernel 2: pair_emb[i,j,:] as gather + AXPY (one-hot GEMM collapsed):
//   W[p] + ent*W[66] + W[67+ch] + b + tfi[i] + tfj[j]
//   Block: fixed i, 64 j's; 8 waves, each wave one j per iter,
//   lane holds 4 channels (32*4 = 128). w_relpos lives in LDS (37 KB).
// ---------------------------------------------------------------------------
#define JT 64

__global__ __launch_bounds__(256) void pair_kernel(
    const int* __restrict__ ri,  const int* __restrict__ sym,
    const int* __restrict__ asym, const int* __restrict__ ent,
    const float* __restrict__ w_rel, const float* __restrict__ b_rel,
    const float* __restrict__ tfi,   const float* __restrict__ tfj,
    float* __restrict__ out)
{
  __shared__ float Ws[NO_BINS * CZ];       // 37,376 B
  const int tid = threadIdx.x;
  for (int idx = tid; idx < NO_BINS * CZ; idx += 256) Ws[idx] = w_rel[idx];

  const int i     = blockIdx.x;
  const int jbase = blockIdx.y * JT;
  const int wave  = tid >> 5;
  const int lane  = tid & 31;
  const int c     = lane * 4;

  const int ri_i = ri[i], sym_i = sym[i], asym_i = asym[i], ent_i = ent[i];
  f32x4 base = ld4(tfi + (long)i * CZ + c) + ld4(b_rel + c);

  __syncthreads();

  float* orow = out + ((long)i * NRES + jbase) * CZ;
  const f32x4 went = ld4(&Ws[66 * CZ + c]);   // entity scalar-feature column

  for (int jj = wave; jj < JT; jj += 8) {
    const int j  = jbase + jj;
    const int rj = ri[j], sj = sym[j], aj = asym[j];

    int off = ri_i - rj + MAXR;
    off = min(max(off, 0), 2 * MAXR);
    const int p = (asym_i == aj) ? off : (2 * MAXR + 1);     // 0..65

    const int e = ent_i - sj;
    int ch = min(max(sym_i - sj + MAXC, 0), 2 * MAXC);
    ch = (e != 0) ? ch : (2 * MAXC + 1);                     // 0..5
    const float ef = (float)e;

    f32x4 r = base;
    r += ld4(&Ws[p * CZ + c]);
    r += ef * went;
    r += ld4(&Ws[(67 + ch) * CZ + c]);
    r += ld4(tfj + (long)j * CZ + c);
    st4nt(orow + (long)jj * CZ + c, r);
  }
}

// ---------------------------------------------------------------------------
extern "C" void kernel_launch(void* const* d_in, const int* in_sizes, int n_in,
                              void* d_out, int out_size, void* d_ws, size_t ws_size,
                              hipStream_t stream) {
  const float* target_feat = (const float*)d_in[0];
  const float* msa_feat    = (const float*)d_in[1];
  const int*   residue_idx = (const int*)d_in[2];
  const int*   sym_id      = (const int*)d_in[3];
  const int*   asym_id     = (const int*)d_in[4];
  const int*   entity_id   = (const int*)d_in[5];
  const float* w_tf_z_i    = (const float*)d_in[6];
  const float* b_tf_z_i    = (const float*)d_in[7];
  const float* w_tf_z_j    = (const float*)d_in[8];
  const float* b_tf_z_j    = (const float*)d_in[9];
  const float* w_tf_m      = (const float*)d_in[10];
  const float* b_tf_m      = (const float*)d_in[11];
  const float* w_msa_m     = (const float*)d_in[12];
  const float* b_msa_m     = (const float*)d_in[13];
  const float* w_relpos    = (const float*)d_in[14];
  const float* b_relpos    = (const float*)d_in[15];

  float* out = (float*)d_out;
  float* ws  = (float*)d_ws;

  float* tfi = ws;                         // 768*128
  float* tfj = ws + NRES * CZ;             // 768*128
  float* tfm = ws + 2 * NRES * CZ;         // 768*256

  // 0) tiny projections into workspace
  tf_proj_kernel<<<(NRES * 512) / 256, 256, 0, stream>>>(
      target_feat, w_tf_z_i, b_tf_z_i, w_tf_z_j, b_tf_z_j, w_tf_m, b_tf_m, ws);

  // 1) msa_emb via fp32 WMMA: 393216 rows / 128 rows-per-block
  msa_wmma_kernel<<<(SROW * NRES) / (MT * 16), 256, 0, stream>>>(
      msa_feat, w_msa_m, b_msa_m, tfm, out);

  // 2) pair_emb (starts after msa_emb = S*N*CM floats)
  float* pair_out = out + (long)SROW * NRES * CM;
  pair_kernel<<<dim3(NRES, NRES / JT), 256, 0, stream>>>(
      residue_idx, sym_id, asym_id, entity_id, w_relpos, b_relpos,
      tfi, tfj, pair_out);
}